// Displacer_30485677867845
// MI455X (gfx1250) — compile-verified
//
#include <hip/hip_runtime.h>
#include <stdint.h>

typedef __attribute__((ext_vector_type(2))) float v2f;
typedef __attribute__((ext_vector_type(8))) float v8f;
typedef __attribute__((ext_vector_type(4))) unsigned int u32x4;
typedef __attribute__((ext_vector_type(8))) int i32x8;
typedef __attribute__((ext_vector_type(4))) int i32x4;

#define NN   1024
#define BB   16
#define BM   128
#define BN   64
#define KC   32
#define LDST 36       // LDS row stride in floats = 32 data + 4 pad (TDM pad reproduces this)
#define NCH  (NN / KC)

#if defined(__has_builtin)
#if __has_builtin(__builtin_amdgcn_tensor_load_to_lds) && \
    __has_builtin(__builtin_amdgcn_s_wait_tensorcnt)
#define HAVE_TDM 1
#endif
#endif
#ifndef HAVE_TDM
#define HAVE_TDM 0
#endif

#if HAVE_TDM
typedef __attribute__((address_space(3))) void lds_void_t;

static __device__ __forceinline__ uint32_t lds_byte_offset(const void* p) {
    return (uint32_t)(uintptr_t)(lds_void_t*)(p);
}

// Issue one TDM 2D tile load: rows x KC f32 tile from row-major E (row stride NN)
// into LDS at lds_off, with 4-DWORD padding after every 32 DWORDs (row stride 36 floats).
// D# layout per CDNA5 ISA.
// amdgpu-toolchain (clang-23) 6-arg builtin form: (g0, g1, g2, g3, g4, cpol).
static __device__ __forceinline__ void tdm_load_tile(const float* gsrc,
                                                     uint32_t lds_off, int rows) {
    uint64_t ga = (uint64_t)(uintptr_t)gsrc;

    u32x4 g0;
    g0[0] = 1u;                                             // count=1, user descriptor
    g0[1] = lds_off;                                        // lds_addr (bytes)
    g0[2] = (uint32_t)ga;                                   // global_addr[31:0]
    g0[3] = (uint32_t)((ga >> 32) & 0x01FFFFFFu) | (2u << 30); // addr[56:32] | type=2

    // group1 bitfields packed via four 64-bit words (little-endian dword order)
    uint64_t q0 = ((uint64_t)2u << 16)                      // data_size = 4 bytes
                | ((uint64_t)1u << 20)                      // pad_enable
                | ((uint64_t)4u << 22)                      // pad_interval: 32 DWORDs
                | ((uint64_t)3u << 25)                      // pad_amount: 4 DWORDs
                | ((uint64_t)(NN & 0xFFFF) << 48);          // tensor_dim0[15:0]
    uint64_t q1 = (uint64_t)(NN >> 16)                      // tensor_dim0[31:16]
                | ((uint64_t)NN << 16)                      // tensor_dim1
                | ((uint64_t)KC << 48);                     // tile_dim0 = 32 (k)
    uint64_t q2 = (uint64_t)(uint32_t)rows                  // tile_dim1 = rows
                | ((uint64_t)(uint32_t)NN << 32);           // tensor_dim0_stride[31:0]
    uint64_t q3 = ((uint64_t)NN << 16);                     // tensor_dim1_stride (unused 2D)

    i32x8 g1 = { (int)(uint32_t)q0, (int)(q0 >> 32),
                 (int)(uint32_t)q1, (int)(q1 >> 32),
                 (int)(uint32_t)q2, (int)(q2 >> 32),
                 (int)(uint32_t)q3, (int)(q3 >> 32) };
    i32x4 gz4 = {0, 0, 0, 0};                               // groups 2/3: dims 3..5 unused
    i32x8 gz8 = {0, 0, 0, 0, 0, 0, 0, 0};                   // group 4: unused

    __builtin_amdgcn_tensor_load_to_lds(g0, g1, gz4, gz4, gz8, 0);
}
#endif  // HAVE_TDM

__launch_bounds__(256)
__global__ void displacer_wmma_f32(const float* __restrict__ aR,
                                   const float* __restrict__ aI,
                                   const float* __restrict__ evals,
                                   const float* __restrict__ E,
                                   float* __restrict__ out)
{
    __shared__ float As[2][BM][LDST];   // double-buffered E rows, M strip
    __shared__ float Bs[2][BN][LDST];   // double-buffered E rows, N strip (B = E^T)
    __shared__ float csb[2][KC];        // cos(r*lambda)
    __shared__ float ssb[2][KC];        // -sin(r*lambda)

    const int tid  = threadIdx.x;
    const int lane = tid & 31;
    const int wave = tid >> 5;
    const int lh   = lane & 15;         // M (A) / N (B) index within fragment
    const int kg   = (lane >> 4) * 2;   // K pair base within 4-wide K step

    const int b  = blockIdx.z;
    const int m0 = blockIdx.y * BM;
    const int n0 = blockIdx.x * BN;

    const float ar = aR[b];
    const float ai = aI[b];
    const float r  = sqrtf(ar * ar + ai * ai) + 1e-10f;

    const v8f vzero = {0.f, 0.f, 0.f, 0.f, 0.f, 0.f, 0.f, 0.f};
    v8f accR[4], accI[4];
#pragma unroll
    for (int t = 0; t < 4; ++t) { accR[t] = vzero; accI[t] = vzero; }

    auto stage = [&](int k0, int buf) {
#if HAVE_TDM
        if (wave == 0) {
            tdm_load_tile(E + (size_t)m0 * NN + k0, lds_byte_offset(&As[buf][0][0]), BM);
            tdm_load_tile(E + (size_t)n0 * NN + k0, lds_byte_offset(&Bs[buf][0][0]), BN);
        }
#else
#pragma unroll
        for (int q = 0; q < 4; ++q) {
            int idx = tid + 256 * q;
            int row = idx >> 3;
            int c4  = (idx & 7) * 4;
            float4 v = *(const float4*)(E + (size_t)(m0 + row) * NN + k0 + c4);
            *(float4*)(&As[buf][row][c4]) = v;
        }
#pragma unroll
        for (int q = 0; q < 2; ++q) {
            int idx = tid + 256 * q;
            int row = idx >> 3;
            int c4  = (idx & 7) * 4;
            float4 v = *(const float4*)(E + (size_t)(n0 + row) * NN + k0 + c4);
            *(float4*)(&Bs[buf][row][c4]) = v;
        }
#endif
        if (tid < KC) {                 // exp(-i r lambda) diagonal chunk
            float s, c;
            __sincosf(r * evals[k0 + tid], &s, &c);
            csb[buf][tid] = c;
            ssb[buf][tid] = -s;
        }
    };

    stage(0, 0);                        // prologue: chunk 0 in flight

    for (int c = 0; c < NCH; ++c) {
        const int buf = c & 1;
        if (c + 1 < NCH) stage((c + 1) * KC, buf ^ 1);  // pipeline next chunk
#if HAVE_TDM
        if (wave == 0) {
            if (c + 1 < NCH) __builtin_amdgcn_s_wait_tensorcnt(2);  // chunk c done
            else             __builtin_amdgcn_s_wait_tensorcnt(0);
        }
#endif
        __syncthreads();                // chunk c tiles + cs/ss visible to all waves

        // 8 K-steps of 4; per step: 1 A frag (scaled twice) x 4 shared B frags -> 8 WMMAs
#pragma unroll
        for (int kk = 0; kk < KC; kk += 4) {
            float2 ae = *(const float2*)(&As[buf][wave * 16 + lh][kk + kg]);
            float2 cp = *(const float2*)(&csb[buf][kk + kg]);
            float2 sp = *(const float2*)(&ssb[buf][kk + kg]);
            v2f aRe = {ae.x * cp.x, ae.y * cp.y};
            v2f aIm = {ae.x * sp.x, ae.y * sp.y};
#pragma unroll
            for (int t = 0; t < 4; ++t) {
                float2 be = *(const float2*)(&Bs[buf][t * 16 + lh][kk + kg]);
                v2f bf = {be.x, be.y};
                accR[t] = __builtin_amdgcn_wmma_f32_16x16x4_f32(
                    false, aRe, false, bf, (short)0, accR[t], false, false);
                accI[t] = __builtin_amdgcn_wmma_f32_16x16x4_f32(
                    false, aIm, false, bf, (short)0, accI[t], false, false);
            }
        }
        __syncthreads();                // buf free for reuse by chunk c+2
    }

    // epilogue: Da[i,j] = e^{i (i-j) theta} * (S_re + i S_im), phi = i*alpha/r
    const float theta = atan2f(ar, -ai);
    const int   iB    = m0 + wave * 16 + 8 * (lane >> 4);
#pragma unroll
    for (int t = 0; t < 4; ++t) {
        const int j = n0 + t * 16 + lh;
#pragma unroll
        for (int g = 0; g < 8; ++g) {
            int    i   = iB + g;
            double ang = (double)(i - j) * (double)theta;
            double f   = ang * 0.15915494309189535;   // /(2*pi)
            f -= rint(f);
            float a2 = (float)(f * 6.283185307179586);
            float cr = cosf(a2);
            float sr = sinf(a2);
            float re = accR[t][g];
            float im = accI[t][g];
            float2 o;
            o.x = cr * re - sr * im;
            o.y = sr * re + cr * im;
            *(float2*)(out + 2 * (((size_t)b * NN + i) * NN + j)) = o;
        }
    }
}

extern "C" void kernel_launch(void* const* d_in, const int* in_sizes, int n_in,
                              void* d_out, int out_size, void* d_ws, size_t ws_size,
                              hipStream_t stream) {
    (void)in_sizes; (void)n_in; (void)d_ws; (void)ws_size; (void)out_size;
    const float* aR    = (const float*)d_in[0];  // alpha_real [16]
    const float* aI    = (const float*)d_in[1];  // alpha_imag [16]
    const float* evals = (const float*)d_in[2];  // [1024]
    const float* evecs = (const float*)d_in[3];  // [1024,1024] row-major E[i,k]
    float* out = (float*)d_out;                  // [16,1024,1024] complex64 as float pairs

    dim3 grid(NN / BN, NN / BM, BB);             // (16, 8, 16)
    dim3 block(256);
    displacer_wmma_f32<<<grid, block, 0, stream>>>(aR, aI, evals, evecs, out);
}